// LocalFusionModule_refindices_9620726743505
// MI455X (gfx1250) — compile-verified
//
#include <hip/hip_runtime.h>

typedef __attribute__((ext_vector_type(2))) float v2f;
typedef __attribute__((ext_vector_type(8))) float v8f;

#define B_   8
#define N_   4
#define NREF 3
#define C_   256
#define HW_  4096
#define L_   2048

// ---------------------------------------------------------------------------
// 1) copy feat -> out (vectorized float4). 8*256*4096 floats = 2,097,152 f4.
// ---------------------------------------------------------------------------
__global__ __launch_bounds__(256) void copy_feat_kernel(const float4* __restrict__ src,
                                                        float4* __restrict__ dst) {
  size_t i = (size_t)blockIdx.x * blockDim.x + threadIdx.x;
  dst[i] = src[i];
}

// ---------------------------------------------------------------------------
// 2) gather + l2-normalize feat columns -> wsA  (b, L, c) row-major
//    one block per (b,l); 256 threads = channels
// ---------------------------------------------------------------------------
__global__ __launch_bounds__(256) void prep_feat_kernel(const float* __restrict__ feat,
                                                        const int* __restrict__ base_idx,
                                                        float* __restrict__ wsA) {
  const int bl = blockIdx.x;            // b*L + l
  const int b  = bl >> 11;
  const int ch = threadIdx.x;
  const int col = base_idx[bl];

  float v = feat[((size_t)(b * C_ + ch)) * HW_ + col];

  float s = v * v;
#pragma unroll
  for (int off = 16; off >= 1; off >>= 1) s += __shfl_xor(s, off, 32);
  __shared__ float wsum[8];
  if ((threadIdx.x & 31) == 0) wsum[threadIdx.x >> 5] = s;
  __syncthreads();
  float tot = 0.f;
#pragma unroll
  for (int i = 0; i < 8; ++i) tot += wsum[i];

  float denom = fmaxf(sqrtf(tot), 1e-12f);
  wsA[(size_t)bl * C_ + ch] = v / denom;
}

// ---------------------------------------------------------------------------
// 3) gather + l2-normalize ref columns -> wsB  (b, n, L, c) row-major
//    one block per (b,n,l); 256 threads = channels
// ---------------------------------------------------------------------------
__global__ __launch_bounds__(256) void prep_refs_kernel(const float* __restrict__ refs,
                                                        const int* __restrict__ ref_idx,
                                                        const int* __restrict__ d_index,
                                                        float* __restrict__ wsB) {
  const int bnl = blockIdx.x;           // b*NREF*L + nn*L + l
  const int b   = bnl / (NREF * L_);
  const int rem = bnl - b * (NREF * L_);
  const int nn  = rem >> 11;
  const int ch  = threadIdx.x;
  const int idx = *d_index;
  const int slot = (nn < idx) ? nn : nn + 1;   // skip excluded ref slice
  const int col  = ref_idx[bnl];

  float v = refs[(((size_t)(b * N_ + slot)) * C_ + ch) * HW_ + col];

  float s = v * v;
#pragma unroll
  for (int off = 16; off >= 1; off >>= 1) s += __shfl_xor(s, off, 32);
  __shared__ float wsum[8];
  if ((threadIdx.x & 31) == 0) wsum[threadIdx.x >> 5] = s;
  __syncthreads();
  float tot = 0.f;
#pragma unroll
  for (int i = 0; i < 8; ++i) tot += wsum[i];

  float denom = fmaxf(sqrtf(tot), 1e-12f);
  wsB[(size_t)bnl * C_ + ch] = v / denom;
}

// ---------------------------------------------------------------------------
// 4) WMMA f32 GEMM (16x16 tile per wave, K=256) fused with row-argmax.
//    grid = 24 (b,n) * 128 row-tiles = 3072 one-wave blocks.
//    A fragment: lanes 0-15 row M (K pair k,k+1); lanes 16-31 row M (k+2,k+3)
//    B fragment: lanes 0-15 col N (K pair k,k+1); lanes 16-31 col N (k+2,k+3)
//    C layout:   vgpr r -> row r (lanes 0-15), row r+8 (lanes 16-31)
// ---------------------------------------------------------------------------
__global__ __launch_bounds__(32) void gemm_argmax_kernel(const float* __restrict__ wsA,
                                                         const float* __restrict__ wsB,
                                                         int* __restrict__ wsArg) {
  __shared__ float alds[16 * 258];       // padded stride: avoid 64-bank conflicts
  const int lane = threadIdx.x;
  const int bn   = blockIdx.x >> 7;      // 0..23  (b*NREF + nn)
  const int tile = blockIdx.x & 127;     // row tile
  const int l0   = tile << 4;
  const int b    = bn / NREF;
  const int sub  = lane & 15;
  const int half = lane >> 4;

  // stage 16x256 A tile (contiguous rows of wsA) into LDS
  const float* Ab = wsA + ((size_t)b * L_ + l0) * C_;
  for (int i = lane; i < 16 * C_; i += 32)
    alds[(i >> 8) * 258 + (i & 255)] = Ab[i];
  __syncthreads();

  const float* aptr  = alds + sub * 258 + 2 * half;
  const float* Bbase = wsB + (size_t)bn * L_ * C_;

  float best[8];
  int   bidx[8];
#pragma unroll
  for (int r = 0; r < 8; ++r) { best[r] = -3.0e38f; bidx[r] = 0; }

  for (int j0 = 0; j0 < L_; j0 += 32) {
    v8f acc0 = {};
    v8f acc1 = {};
    const float* bptr0 = Bbase + ((size_t)(j0 + sub)) * C_ + 2 * half;
    const float* bptr1 = bptr0 + 16 * C_;
#pragma unroll 8
    for (int k = 0; k < C_; k += 4) {
      v2f a  = *(const v2f*)(aptr + k);
      v2f b0 = *(const v2f*)(bptr0 + k);
      v2f b1 = *(const v2f*)(bptr1 + k);
      acc0 = __builtin_amdgcn_wmma_f32_16x16x4_f32(false, a, false, b0,
                                                   (short)0, acc0, false, false);
      acc1 = __builtin_amdgcn_wmma_f32_16x16x4_f32(false, a, false, b1,
                                                   (short)0, acc1, false, false);
    }
    // running per-lane max/argmax (ascending j => strict '>' keeps first max)
#pragma unroll
    for (int r = 0; r < 8; ++r) {
      float v0 = acc0[r];
      bool g0 = v0 > best[r];
      best[r] = g0 ? v0 : best[r];
      bidx[r] = g0 ? (j0 + sub) : bidx[r];
      float v1 = acc1[r];
      bool g1 = v1 > best[r];
      best[r] = g1 ? v1 : best[r];
      bidx[r] = g1 ? (j0 + 16 + sub) : bidx[r];
    }
  }

  // reduce across the 16 lanes holding one row; tie-break to lowest index
#pragma unroll
  for (int r = 0; r < 8; ++r) {
    float bv = best[r];
    int   bi = bidx[r];
#pragma unroll
    for (int off = 8; off >= 1; off >>= 1) {
      float ov = __shfl_xor(bv, off, 32);
      int   oi = __shfl_xor(bi, off, 32);
      if (ov > bv || (ov == bv && oi < bi)) { bv = ov; bi = oi; }
    }
    if (sub == 0)                                    // lanes 0 and 16 write
      wsArg[(size_t)bn * L_ + l0 + r + 8 * half] = bi;
  }
}

// ---------------------------------------------------------------------------
// 5) fuse + scatter into out; also emit base_indices / ref_indices as f32
// ---------------------------------------------------------------------------
__global__ __launch_bounds__(256) void fuse_scatter_kernel(const float* __restrict__ feat,
                                                           const float* __restrict__ similarity,
                                                           const int* __restrict__ base_idx,
                                                           const int* __restrict__ wsArg,
                                                           const float* __restrict__ wsB,
                                                           const int* __restrict__ d_index,
                                                           float* __restrict__ out,
                                                           float* __restrict__ out_bidx,
                                                           float* __restrict__ out_ridx) {
  const int bl = blockIdx.x;            // b*L + l
  const int b  = bl >> 11;
  const int l  = bl & 2047;
  const int ch = threadIdx.x;
  const int idx = *d_index;
  const int col = base_idx[bl];

  const float base_sim = similarity[b * N_ + idx];
  float acc = base_sim * feat[((size_t)(b * C_ + ch)) * HW_ + col];

#pragma unroll
  for (int nn = 0; nn < NREF; ++nn) {
    const int slot = (nn < idx) ? nn : nn + 1;
    const float sim = similarity[b * N_ + slot];
    const int arg = wsArg[((size_t)(b * NREF + nn)) * L_ + l];
    acc += sim * wsB[(((size_t)(b * NREF + nn)) * L_ + arg) * C_ + ch];
  }
  out[((size_t)(b * C_ + ch)) * HW_ + col] = acc;

  if (ch == 0) out_bidx[bl] = (float)col;
  if (ch < NREF)
    out_ridx[((size_t)b * NREF + ch) * L_ + l] =
        (float)wsArg[((size_t)(b * NREF + ch)) * L_ + l];
}

// ---------------------------------------------------------------------------
extern "C" void kernel_launch(void* const* d_in, const int* in_sizes, int n_in,
                              void* d_out, int out_size, void* d_ws, size_t ws_size,
                              hipStream_t stream) {
  const float* feat        = (const float*)d_in[0];
  const float* refs        = (const float*)d_in[1];
  const float* similarity  = (const float*)d_in[2];
  const int*   base_idx    = (const int*)d_in[3];
  const int*   ref_idx     = (const int*)d_in[4];
  const int*   d_index     = (const int*)d_in[5];

  float* out      = (float*)d_out;
  float* out_bidx = out + (size_t)B_ * C_ * HW_;          // 8,388,608
  float* out_ridx = out_bidx + (size_t)B_ * L_;           // +16,384

  float* wsA   = (float*)d_ws;                            // b*L*c      = 16.8 MB
  float* wsB   = wsA + (size_t)B_ * L_ * C_;              // b*n*L*c    = 50.3 MB
  int*   wsArg = (int*)(wsB + (size_t)B_ * NREF * L_ * C_);

  copy_feat_kernel<<<(B_ * C_ * HW_) / 4 / 256, 256, 0, stream>>>(
      (const float4*)feat, (float4*)out);
  prep_feat_kernel<<<B_ * L_, 256, 0, stream>>>(feat, base_idx, wsA);
  prep_refs_kernel<<<B_ * NREF * L_, 256, 0, stream>>>(refs, ref_idx, d_index, wsB);
  gemm_argmax_kernel<<<(B_ * NREF) * (L_ / 16), 32, 0, stream>>>(wsA, wsB, wsArg);
  fuse_scatter_kernel<<<B_ * L_, 256, 0, stream>>>(feat, similarity, base_idx, wsArg,
                                                   wsB, d_index, out, out_bidx, out_ridx);
}